// Voxel3DObservability_41704132444296
// MI455X (gfx1250) — compile-verified
//
#include <hip/hip_runtime.h>
#include <stdint.h>

typedef float v4f __attribute__((ext_vector_type(4)));

#define BATCH 4
#define WDIM 160
#define LDIM 160
#define HDIM 80
#define IW 512
#define IH 512
#define NVOX (WDIM * LDIM * HDIM)          // 2,048,000 voxels per batch
#define VPT 8                              // voxels per thread (along H; 80 % 8 == 0)
#define TPB 256
#define THREADS_PER_BATCH (NVOX / VPT)     // 256,000
#define BLOCKS_PER_BATCH (THREADS_PER_BATCH / TPB)  // 1000
#define VSZ 0.05f
#define TOL (VSZ * 0.5f)

__global__ __launch_bounds__(TPB) void Voxel3DObservability_kernel(
    const float* __restrict__ voxel,   // (B,3,W,L,H)
    const float* __restrict__ extr,    // (B,4,4)
    const float* __restrict__ depth,   // (B,1,IH,IW)
    const float* __restrict__ origin,  // (3)
    const int*   __restrict__ hfov,    // (1)
    float* __restrict__ out_mask,      // (B,1,W,L,H)
    float* __restrict__ out_ray)       // (B,1,W,L,H)
{
    // per-thread 32B slot per channel, contiguous -> two B128 async loads each
    __shared__ v4f sbuf[3][TPB][2];    // 24 KB staging

    const int tid  = threadIdx.x;
    const int blk  = blockIdx.x;
    const int b    = blk / BLOCKS_PER_BATCH;                 // block-uniform
    const int r    = (blk - b * BLOCKS_PER_BATCH) * TPB + tid;
    const int idx0 = r * VPT;                                // flat voxel idx in batch

    // ---- kick off async DMA of this thread's 3 channel slices into LDS ----
    const float* vbase = voxel + ((size_t)b * 3u) * (size_t)NVOX + (size_t)idx0;
#pragma unroll
    for (int c = 0; c < 3; ++c) {
        unsigned lds0 = (unsigned)(uintptr_t)&sbuf[c][tid][0];
        unsigned lds1 = (unsigned)(uintptr_t)&sbuf[c][tid][1];
        unsigned long long ga =
            (unsigned long long)(uintptr_t)(vbase + (size_t)c * (size_t)NVOX);
        asm volatile("global_load_async_to_lds_b128 %0, %1, off"
                     :: "v"(lds0), "v"(ga) : "memory");
        asm volatile("global_load_async_to_lds_b128 %0, %1, off offset:16"
                     :: "v"(lds1), "v"(ga) : "memory");
    }

    // ---- overlap: voxel center, camera transform, projection, depth gather ----
    const int x  = idx0 / (LDIM * HDIM);
    const int rm = idx0 - x * (LDIM * HDIM);
    const int y  = rm / HDIM;
    const int z0 = rm - y * HDIM;

    const float ox = origin[0], oy = origin[1], oz = origin[2];
    const float* E = extr + b * 16;    // block-uniform -> scalar loads
    const float r00 = E[0], r01 = E[1], r02 = E[2],  t0 = E[3];
    const float r10 = E[4], r11 = E[5], r12 = E[6],  t1 = E[7];
    const float r20 = E[8], r21 = E[9], r22 = E[10], t2 = E[11];

    const float px = ((float)x  + 0.5f) * VSZ + ox;
    const float py = ((float)y  + 0.5f) * VSZ + oy;
    const float pz = ((float)z0 + 0.5f) * VSZ + oz;

    float cx = r00 * px + r01 * py + r02 * pz + t0;
    float cy = r10 * px + r11 * py + r12 * pz + t1;
    float cz = r20 * px + r21 * py + r22 * pz + t2;
    const float dcx = r02 * VSZ, dcy = r12 * VSZ, dcz = r22 * VSZ;

    // f = (iw/2) / tan(deg2rad(hfov)/2)   (fast tan: arg is ~pi/4, well in range)
    const float f = ((float)IW * 0.5f) /
                    __tanf((float)hfov[0] * (3.14159265358979323846f / 360.0f));
    const float* dimg = depth + (size_t)b * (size_t)(IH * IW);

    v4f rayv[2];
    int  obs[VPT];
#pragma unroll
    for (int k = 0; k < VPT; ++k) {
        const float w  = f * __builtin_amdgcn_rcpf(cz);   // v_rcp_f32 (1-ulp)
        const float u  = cx * w + 0.5f * (float)IW;
        const float v  = cy * w + 0.5f * (float)IH;
        const bool inb = (u > 0.0f) && (u < (float)IH) &&
                         (v > 0.0f) && (v < (float)IW) && (cz > 0.0f);
        const int ui = inb ? (int)u : 0;
        const int vi = inb ? (int)v : 0;
        const float rv = dimg[vi * IW + ui];   // L2-resident gather (4 MB footprint)
        rayv[k >> 2][k & 3] = rv;
        obs[k] = inb && (cz <= rv + TOL);
        cx += dcx; cy += dcy; cz += dcz;
    }

    // ---- wait for DMA, fold occupancy, emit mask ----
    asm volatile("s_wait_asynccnt 0x0" ::: "memory");
    v4f m[2];
#pragma unroll
    for (int h = 0; h < 2; ++h) {
        const v4f c0v = sbuf[0][tid][h];
        const v4f c1v = sbuf[1][tid][h];
        const v4f c2v = sbuf[2][tid][h];
#pragma unroll
        for (int j = 0; j < 4; ++j) {
            const float mx = fmaxf(c0v[j], fmaxf(c1v[j], c2v[j]));
            m[h][j] = (obs[h * 4 + j] || (mx > 0.2f)) ? 1.0f : 0.0f;
        }
    }

    // streamed outputs: non-temporal 128-bit stores
    const size_t t2q = ((size_t)blk * TPB + (size_t)tid) * 2;  // in v4f units
    __builtin_nontemporal_store(m[0],    ((v4f*)out_mask) + t2q);
    __builtin_nontemporal_store(m[1],    ((v4f*)out_mask) + t2q + 1);
    __builtin_nontemporal_store(rayv[0], ((v4f*)out_ray)  + t2q);
    __builtin_nontemporal_store(rayv[1], ((v4f*)out_ray)  + t2q + 1);
}

extern "C" void kernel_launch(void* const* d_in, const int* in_sizes, int n_in,
                              void* d_out, int out_size, void* d_ws, size_t ws_size,
                              hipStream_t stream) {
    const float* voxel  = (const float*)d_in[0];
    const float* extr   = (const float*)d_in[1];
    const float* depth  = (const float*)d_in[2];
    const float* origin = (const float*)d_in[3];
    const int*   hfov   = (const int*)d_in[4];

    float* out_mask = (float*)d_out;
    float* out_ray  = out_mask + (size_t)BATCH * (size_t)NVOX;

    dim3 grid(BATCH * BLOCKS_PER_BATCH);   // 4000 blocks, no tail
    Voxel3DObservability_kernel<<<grid, dim3(TPB), 0, stream>>>(
        voxel, extr, depth, origin, hfov, out_mask, out_ray);
}